// GCN_62466004353421
// MI455X (gfx1250) — compile-verified
//
#include <hip/hip_runtime.h>

// CDNA5 / gfx1250, wave32. f32 WMMA path (workload is HBM-bandwidth bound;
// inputs are f32, so V_WMMA_F32_16X16X4_F32 is the right matrix instruction).
// Streamed-once matrices (onehot: 410MB, x: 51MB) are loaded non-temporal so
// the 192MB L2 stays resident with the gather/scatter targets (h buffers).

typedef __attribute__((ext_vector_type(2))) float v2f;
typedef __attribute__((ext_vector_type(8))) float v8f;

#define NN 100000            // N_NODES
#define NE 1600000           // N_EDGES
#define NOBS 1024            // onehot rows
#define SPLITK 50            // split-K chunks for final GEMM (N/SPLITK = 2000)

// ---------------------------------------------------------------------------
// Degree / symmetric-norm kernels
// ---------------------------------------------------------------------------
__global__ void k_deg_init(float* __restrict__ deg) {
  int n = blockIdx.x * blockDim.x + threadIdx.x;
  if (n < NN) deg[n] = 1.0f;                       // self-loop contribution
}

__global__ void k_deg_acc(const int* __restrict__ dst, float* __restrict__ deg) {
  int e = blockIdx.x * blockDim.x + threadIdx.x;
  if (e < NE) unsafeAtomicAdd(&deg[dst[e]], 1.0f); // global_atomic_add_f32
}

__global__ void k_dinv(float* __restrict__ deg) {  // in-place deg -> deg^-1/2
  int n = blockIdx.x * blockDim.x + threadIdx.x;
  if (n < NN) {
    float v = deg[n];
    deg[n] = (v > 0.0f) ? rsqrtf(v) : 0.0f;
  }
}

// ---------------------------------------------------------------------------
// Node-feature GEMM:  G[n, c] = sum_k A(n,k) * W[k, c],  K = 128.
// TRANS_A=1: A(n,k) = Abuf[k*NN + n]  (x stored [128, N] row-major; NT loads)
// TRANS_A=0: A(n,k) = Abuf[n*128 + k] (a1 stored [N, 128] row-major; b64 loads)
// One wave per 16x16 output tile; NOUT/16 waves per block; blockIdx.x = m-tile.
// ---------------------------------------------------------------------------
template <int NOUT, bool RELU, bool TRANS_A>
__global__ void k_gemm(const float* __restrict__ A, const float* __restrict__ W,
                       float* __restrict__ G) {
  const int wave  = threadIdx.x >> 5;     // n-tile within block
  const int lane  = threadIdx.x & 31;
  const int mt    = blockIdx.x;           // node tile (16 nodes)
  const int l16   = lane & 15;
  const int hi    = lane >> 4;            // 0 | 1 (K split per ISA A/B layout)
  const int node  = mt * 16 + l16;        // A-frag: M = lane%16
  const int col   = wave * 16 + l16;      // B-frag: N = lane%16
  v8f c = {0.f, 0.f, 0.f, 0.f, 0.f, 0.f, 0.f, 0.f};

  for (int k0 = 0; k0 < 128; k0 += 4) {
    const int ka = k0 + hi * 2;           // lanes 0-15: K={k0,k0+1}; 16-31: +2
    v2f a, b;
    if (TRANS_A) {
      // column access of x: strided, streamed once -> non-temporal
      a.x = __builtin_nontemporal_load(A + (size_t)ka * NN + node);
      a.y = __builtin_nontemporal_load(A + (size_t)(ka + 1) * NN + node);
    } else {
      // K-consecutive, 8B aligned (128*node + ka, ka even) -> single b64
      a = *(const v2f*)(A + (size_t)node * 128 + ka);
    }
    if (RELU) { a.x = fmaxf(a.x, 0.0f); a.y = fmaxf(a.y, 0.0f); }
    b.x = W[(size_t)ka * NOUT + col];
    b.y = W[(size_t)(ka + 1) * NOUT + col];
    c = __builtin_amdgcn_wmma_f32_16x16x4_f32(false, a, false, b,
                                              (short)0, c, false, false);
  }
  // C/D layout: VGPR v -> M = v + (lane/16)*8, N = lane%16
  const int mbase = mt * 16 + hi * 8;
#pragma unroll
  for (int v = 0; v < 8; ++v)
    G[(size_t)(mbase + v) * NOUT + col] = c[v];
}

// ---------------------------------------------------------------------------
// Aggregation: out = segment_sum(msg) + self-loop + bias
// ---------------------------------------------------------------------------
template <int F>
__global__ void k_agg_init(const float* __restrict__ G, const float* __restrict__ dinv,
                           const float* __restrict__ bias, float* __restrict__ out) {
  size_t i = (size_t)blockIdx.x * blockDim.x + threadIdx.x;   // float4 index
  if (i < (size_t)NN * (F / 4)) {
    int n  = (int)(i / (F / 4));
    int f4 = (int)(i % (F / 4)) * 4;
    float di = dinv[n];
    float s  = di * di;                  // self-loop norm = dinv[n]^2
    const float4 g = *(const float4*)(G + i * 4);
    float4 o;
    o.x = g.x * s + bias[f4 + 0];
    o.y = g.y * s + bias[f4 + 1];
    o.z = g.z * s + bias[f4 + 2];
    o.w = g.w * s + bias[f4 + 3];
    *(float4*)(out + i * 4) = o;
  }
}

template <int F>
__global__ void k_agg_edges(const float* __restrict__ G, const float* __restrict__ dinv,
                            const int* __restrict__ src, const int* __restrict__ dst,
                            float* __restrict__ out) {
  const int CH = F / 4;                  // 4 features per thread
  long long t = (long long)blockIdx.x * blockDim.x + threadIdx.x;
  if (t >= (long long)NE * CH) return;
  int e  = (int)(t / CH);
  int f4 = (int)(t % CH) * 4;
  int s = src[e], d = dst[e];
  float norm = dinv[s] * dinv[d];
  const float4 g = *(const float4*)(G + (size_t)s * F + f4);  // global_load_b128 (L2 gather)
  float* od = out + (size_t)d * F + f4;
  unsafeAtomicAdd(od + 0, g.x * norm);
  unsafeAtomicAdd(od + 1, g.y * norm);
  unsafeAtomicAdd(od + 2, g.z * norm);
  unsafeAtomicAdd(od + 3, g.w * norm);
}

// ---------------------------------------------------------------------------
// Final readout: out[1024,64] = onehot[1024,NN] @ h2[NN,64], split-K + atomics
// onehot is streamed once (410MB) -> non-temporal; h2 (25MB) stays in L2.
// ---------------------------------------------------------------------------
__global__ void k_out_zero(float* __restrict__ out) {
  int i = blockIdx.x * blockDim.x + threadIdx.x;
  if (i < NOBS * 64) out[i] = 0.0f;
}

__global__ void k_final(const float* __restrict__ OH, const float* __restrict__ H2,
                        float* __restrict__ out) {
  const int wave = threadIdx.x >> 5;      // j-tile (0..3)
  const int lane = threadIdx.x & 31;
  const int l16  = lane & 15;
  const int hi   = lane >> 4;
  const int mt   = blockIdx.x;            // 64 row tiles
  const int KC   = NN / SPLITK;           // 2000
  const int kbeg = blockIdx.y * KC;
  const int m = mt * 16 + l16;
  const int j = wave * 16 + l16;
  v8f c = {0.f, 0.f, 0.f, 0.f, 0.f, 0.f, 0.f, 0.f};

  for (int k0 = kbeg; k0 < kbeg + KC; k0 += 4) {
    const int ka = k0 + hi * 2;
    // OH row segment: 8B aligned (m*NN + ka, both even) -> one NT b64 load
    v2f a = __builtin_nontemporal_load((const v2f*)(OH + (size_t)m * NN + ka));
    v2f b;
    b.x = H2[(size_t)ka * 64 + j];
    b.y = H2[(size_t)(ka + 1) * 64 + j];
    c = __builtin_amdgcn_wmma_f32_16x16x4_f32(false, a, false, b,
                                              (short)0, c, false, false);
  }
  const int mbase = mt * 16 + hi * 8;
#pragma unroll
  for (int v = 0; v < 8; ++v)
    unsafeAtomicAdd(&out[(size_t)(mbase + v) * 64 + j], c[v]);
}

// ---------------------------------------------------------------------------
extern "C" void kernel_launch(void* const* d_in, const int* in_sizes, int n_in,
                              void* d_out, int out_size, void* d_ws, size_t ws_size,
                              hipStream_t stream) {
  const float* x  = (const float*)d_in[0];       // [128, NN]
  const int*   ei = (const int*)d_in[1];         // [2, NE]
  const float* oh = (const float*)d_in[2];       // [1024, NN]
  const float* W1 = (const float*)d_in[3];       // [128, 128]
  const float* b1 = (const float*)d_in[4];       // [128]
  const float* W2 = (const float*)d_in[5];       // [128, 64]
  const float* b2 = (const float*)d_in[6];       // [64]
  float* out = (float*)d_out;                    // [1024, 64]

  const int* src = ei;
  const int* dst = ei + NE;

  // Workspace layout (256B aligned): dinv | bufA (G1 then G2) | bufB (a1 then h2)
  char* ws = (char*)d_ws;
  float* dinv = (float*)ws;                                  // NN floats
  float* bufA = (float*)(ws + (((size_t)NN * 4 + 255) & ~(size_t)255) + 256);
  float* bufB = bufA + (size_t)NN * 128;                     // NN*128 floats each

  // 1) degree -> dinv
  k_deg_init<<<(NN + 255) / 256, 256, 0, stream>>>(dinv);
  k_deg_acc<<<(NE + 255) / 256, 256, 0, stream>>>(dst, dinv);
  k_dinv<<<(NN + 255) / 256, 256, 0, stream>>>(dinv);

  // 2) G1 = x^T @ W1   (A transposed-access with NT loads, 8 waves/block)
  k_gemm<128, false, true><<<NN / 16, 256, 0, stream>>>(x, W1, bufA);

  // 3) a1 = scatter(G1) + self + b1
  {
    long long tot = (long long)NN * 32;  // float4 elements
    k_agg_init<128><<<(int)((tot + 255) / 256), 256, 0, stream>>>(bufA, dinv, b1, bufB);
    long long et = (long long)NE * 32;   // E * F/4
    k_agg_edges<128><<<(int)((et + 255) / 256), 256, 0, stream>>>(bufA, dinv, src, dst, bufB);
  }

  // 4) G2 = relu(a1) @ W2  (ReLU folded into A-operand load; reuses bufA)
  k_gemm<64, true, false><<<NN / 16, 128, 0, stream>>>(bufB, W2, bufA);

  // 5) h2 = scatter(G2) + self + b2  (reuses bufB)
  {
    long long tot = (long long)NN * 16;  // float4 elements
    k_agg_init<64><<<(int)((tot + 255) / 256), 256, 0, stream>>>(bufA, dinv, b2, bufB);
    long long et = (long long)NE * 16;
    k_agg_edges<64><<<(int)((et + 255) / 256), 256, 0, stream>>>(bufA, dinv, src, dst, bufB);
  }

  // 6) out = onehot @ h2  (split-K WMMA + f32 atomics; onehot NT-streamed)
  k_out_zero<<<(NOBS * 64 + 255) / 256, 256, 0, stream>>>(out);
  k_final<<<dim3(NOBS / 16, SPLITK), 128, 0, stream>>>(oh, bufB, out);
}